// TopK_609885356663
// MI455X (gfx1250) — compile-verified
//
#include <hip/hip_runtime.h>

#define N_COLS 32768
#define TOPK   32
#define NT     1024        // 32 wave32 waves per workgroup; one workgroup per row
#define HREP   8           // histogram bank-conflict replication factor
#define NBIN   256
#define CAP    4096        // candidate buffer capacity (fallback if exceeded)

typedef unsigned v4u __attribute__((ext_vector_type(4)));
typedef float    v4f __attribute__((ext_vector_type(4)));

// Monotonic float->uint key: larger float  <=>  larger key
__device__ __forceinline__ unsigned bits2key(unsigned u) {
    return u ^ (unsigned)(((int)u >> 31) | (int)0x80000000);
}
__device__ __forceinline__ float key2float(unsigned k) {
    unsigned u = (k & 0x80000000u) ? (k ^ 0x80000000u) : ~k;
    return __uint_as_float(u);
}

extern "C" __global__ __launch_bounds__(NT)
void topk_rows_kernel(const float* __restrict__ x, float* __restrict__ out) {
    extern __shared__ unsigned smem[];
    unsigned* keys = smem;                         // N_COLS words
    unsigned* hist = smem + N_COLS;                // NBIN * HREP words
    unsigned* cand = hist + NBIN * HREP;           // CAP words
    unsigned* ctrl = cand + CAP;                   // [0]=prefix [1]=remK [2]=candCnt [3]=eqCnt

    const unsigned tid = threadIdx.x;
    const unsigned row = blockIdx.x;
    const unsigned rep = tid & (HREP - 1);         // per-thread histogram replica
    const unsigned long long rowPtr =
        (unsigned long long)(x + (size_t)row * N_COLS);

    // -----------------------------------------------------------------
    // Stage the whole 128 KB row into LDS with CDNA5 async copies
    // (GVS: saddr = row base SGPR pair, vaddr = byte offset, vdst = LDS addr).
    // -----------------------------------------------------------------
    const unsigned ldsBase = (unsigned)(size_t)keys;
#pragma unroll
    for (int j = 0; j < (N_COLS / 4) / NT; ++j) {  // 8 x b128 per thread
        unsigned byteOff = (tid + (unsigned)j * NT) * 16u;
        unsigned ldsAddr = ldsBase + byteOff;
        asm volatile("global_load_async_to_lds_b128 %0, %1, %2 offset:0"
                     :
                     : "v"(ldsAddr), "v"(byteOff), "s"(rowPtr)
                     : "memory");
    }
    asm volatile("s_wait_asynccnt 0" ::: "memory");

    // ----- shared helpers -------------------------------------------------
    auto zeroHist = [&]() {
        for (unsigned i = tid; i < NBIN * HREP; i += NT) hist[i] = 0u;
        __syncthreads();
    };
    auto finishPass = [&](int p) {
        // reduce replicas: 256 threads each fold their bin into replica 0
        for (unsigned b = tid; b < (unsigned)NBIN; b += NT) {
            unsigned s = 0;
#pragma unroll
            for (int r = 0; r < HREP; ++r) s += hist[b * HREP + r];
            hist[b * HREP] = s;
        }
        __syncthreads();
        if (tid == 0) {
            unsigned remK = ctrl[1];
            unsigned cum = 0, chosen = 0;
            for (int b = NBIN - 1; b >= 0; --b) {
                unsigned c = hist[(unsigned)b * HREP];
                if (cum + c >= remK) { chosen = (unsigned)b; break; }
                cum += c;
            }
            ctrl[0] = (p == 0) ? chosen : ((ctrl[0] << 8) | chosen);
            ctrl[1] = remK - cum;
        }
        __syncthreads();
    };

    // -----------------------------------------------------------------
    // Pass 0: fused  float->key conversion (in place)  +  top-byte histogram.
    // -----------------------------------------------------------------
    if (tid == 0) { ctrl[1] = TOPK; ctrl[2] = 0; ctrl[3] = 0; }
    zeroHist();    // barrier here also publishes all waves' async LDS writes

    v4u* kv = (v4u*)keys;
#pragma unroll
    for (int j = 0; j < (N_COLS / 4) / NT; ++j) {
        unsigned v = tid + (unsigned)j * NT;
        v4u d = kv[v];
        d.x = bits2key(d.x); d.y = bits2key(d.y);
        d.z = bits2key(d.z); d.w = bits2key(d.w);
        kv[v] = d;
        atomicAdd(&hist[(d.x >> 24) * HREP + rep], 1u);
        atomicAdd(&hist[(d.y >> 24) * HREP + rep], 1u);
        atomicAdd(&hist[(d.z >> 24) * HREP + rep], 1u);
        atomicAdd(&hist[(d.w >> 24) * HREP + rep], 1u);
    }
    __syncthreads();
    finishPass(0);

    // -----------------------------------------------------------------
    // Pass 1: byte-1 histogram over matching elements, fused with
    // compaction of the candidates into cand[] (typically ~1k of 32k).
    // -----------------------------------------------------------------
    zeroHist();
    {
        const unsigned b0 = ctrl[0];
#pragma unroll
        for (int j = 0; j < (N_COLS / 4) / NT; ++j) {
            unsigned v = tid + (unsigned)j * NT;
            v4u d = kv[v];
#pragma unroll
            for (int c = 0; c < 4; ++c) {
                unsigned key = (c == 0) ? d.x : (c == 1) ? d.y : (c == 2) ? d.z : d.w;
                if ((key >> 24) == b0) {
                    atomicAdd(&hist[((key >> 16) & 255u) * HREP + rep], 1u);
                    unsigned pos = atomicAdd(&ctrl[2], 1u);
                    if (pos < CAP) cand[pos] = key;
                }
            }
        }
    }
    __syncthreads();
    finishPass(1);

    // -----------------------------------------------------------------
    // Passes 2/3 on the compacted candidates (fallback: full array).
    // -----------------------------------------------------------------
    const unsigned cnt      = ctrl[2];
    const bool     useCand  = (cnt <= CAP);
    const unsigned* src     = useCand ? cand : keys;
    const unsigned  srcN    = useCand ? cnt : (unsigned)N_COLS;

    zeroHist();
    {
        const unsigned pre16 = ctrl[0];
        for (unsigned i = tid; i < srcN; i += NT) {
            unsigned key = src[i];
            if ((key >> 16) == pre16)
                atomicAdd(&hist[((key >> 8) & 255u) * HREP + rep], 1u);
        }
    }
    __syncthreads();
    finishPass(2);

    zeroHist();
    {
        const unsigned pre24 = ctrl[0];
        for (unsigned i = tid; i < srcN; i += NT) {
            unsigned key = src[i];
            if ((key >> 8) == pre24)
                atomicAdd(&hist[(key & 255u) * HREP + rep], 1u);
        }
    }
    __syncthreads();
    finishPass(3);

    const unsigned tk    = ctrl[0];     // exact key of the K-th largest
    const unsigned numEq = ctrl[1];     // equal-to-threshold elements to keep
    const float    tf    = key2float(tk);

    // -----------------------------------------------------------------
    // Write phase: full-row non-temporal b128 stores (write-once stream).
    // -----------------------------------------------------------------
    v4f* orow = (v4f*)(out + (size_t)row * N_COLS);

    auto pick = [&](unsigned key) -> float {
        if (key > tk) return fmaxf(key2float(key), 0.0f);
        if (key == tk) {
            unsigned pos = atomicAdd(&ctrl[3], 1u);
            return (pos < numEq) ? fmaxf(tf, 0.0f) : 0.0f;
        }
        return 0.0f;
    };

#pragma unroll
    for (int j = 0; j < (N_COLS / 4) / NT; ++j) {
        unsigned v = tid + (unsigned)j * NT;
        v4u d = kv[v];
        v4f o;
        o.x = pick(d.x);
        o.y = pick(d.y);
        o.z = pick(d.z);
        o.w = pick(d.w);
        __builtin_nontemporal_store(o, &orow[v]);
    }
}

extern "C" void kernel_launch(void* const* d_in, const int* in_sizes, int n_in,
                              void* d_out, int out_size, void* d_ws, size_t ws_size,
                              hipStream_t stream) {
    (void)n_in; (void)out_size; (void)d_ws; (void)ws_size;
    const float* x  = (const float*)d_in[0];
    float* out      = (float*)d_out;
    const int  rows = in_sizes[0] / N_COLS;                    // 8192
    const size_t shmem =
        (size_t)(N_COLS + NBIN * HREP + CAP + 16) * sizeof(unsigned); // ~152 KB

    hipLaunchKernelGGL(topk_rows_kernel, dim3(rows), dim3(NT), shmem, stream,
                       x, out);
}